// KMeansCrossAttention_73942156968039
// MI455X (gfx1250) — compile-verified
//
#include <hip/hip_runtime.h>
#include <hip/hip_bf16.h>

// KMeans cross-attention for MI455X (gfx1250, wave32).
// logits = q@k^T (f32 WMMA 16x16x4), per-key argmax over N, scatter-add v, mean.
// Dual accumulator chains (2 query tiles / iteration) for WMMA-pipe ILP.

typedef __attribute__((ext_vector_type(2))) float v2f;
typedef __attribute__((ext_vector_type(8))) float v8f;

#define B_ 2
#define N_ 1024
#define M_ 65536
#define D_ 128
#define KEYS_PER_WAVE 16
#define WAVES_PER_BLOCK 8
#define KEYS_PER_BLOCK (KEYS_PER_WAVE * WAVES_PER_BLOCK)  // 128
#define QROWS 32       // queries staged per iteration (2 WMMA tiles)
#define QT_STRIDE 132  // 128 + 4 pad floats: lane l -> banks 4l..4l+1, lane l+16 -> 4l+2..4l+3

__device__ __forceinline__ void atomic_add_f32(float* p, float x) {
  unsafeAtomicAdd(p, x);  // global_atomic_add_f32
}

__global__ void zero_kernel(float* __restrict__ out, float* __restrict__ counts) {
  int i = blockIdx.x * blockDim.x + threadIdx.x;
  if (i < B_ * N_ * D_) out[i] = 0.0f;
  if (i < B_ * N_) counts[i] = 0.0f;
}

__global__ __launch_bounds__(256, 1) void argmax_scatter_kernel(
    const float* __restrict__ q, const float* __restrict__ k,
    const float* __restrict__ v, float* __restrict__ out,
    float* __restrict__ counts) {
  __shared__ float qtile[QROWS * QT_STRIDE];  // ~16.9 KB

  const int b    = blockIdx.y;
  const int tid  = threadIdx.x;
  const int wave = tid >> 5;
  const int lane = tid & 31;
  const int col  = lane & 15;  // key column within this wave's 16-key tile
  const int hi   = lane >> 4;  // half-wave selector

  const int mbase = blockIdx.x * KEYS_PER_BLOCK + wave * KEYS_PER_WAVE;
  const int m     = mbase + col;

  // B-operand fragments, resident in VGPRs for the whole kernel.
  // Chunk c covers d in [4c, 4c+4); per WMMA f32 layout lane holds
  // K = {2*hi, 2*hi+1}, column N = lane%16 -> k[m, 4c + 2*hi + {0,1}].
  v2f bfrag[32];
  {
    const float* krow = k + ((size_t)b * M_ + m) * D_ + 2 * hi;
#pragma unroll
    for (int c = 0; c < 32; ++c)
      bfrag[c] = *(const v2f*)(krow + 4 * c);
  }

  float bmax = -__builtin_inff();
  int   bidx = 0;

  for (int nt = 0; nt < N_ / QROWS; ++nt) {
    const int nbase = nt * QROWS;
    __syncthreads();  // protect LDS from previous iteration's readers
    {
      // Cooperative stage of 32x128 q tile: 16 floats per thread.
      const int r  = tid >> 3;        // 0..31 (query row in tile)
      const int c0 = (tid & 7) * 16;  // 0..112
      const float* src = q + ((size_t)b * N_ + nbase + r) * D_ + c0;
      float4 x0 = *(const float4*)(src);
      float4 x1 = *(const float4*)(src + 4);
      float4 x2 = *(const float4*)(src + 8);
      float4 x3 = *(const float4*)(src + 12);
      float* dst = &qtile[r * QT_STRIDE + c0];
      *(float4*)(dst)      = x0;
      *(float4*)(dst + 4)  = x1;
      *(float4*)(dst + 8)  = x2;
      *(float4*)(dst + 12) = x3;
    }
    __syncthreads();

    // Two independent logits tiles (queries nbase..+15 and nbase+16..+31),
    // each accumulated over K=128 in 32 chained WMMAs; chains interleaved.
    v8f acc0 = {0.f, 0.f, 0.f, 0.f, 0.f, 0.f, 0.f, 0.f};
    v8f acc1 = {0.f, 0.f, 0.f, 0.f, 0.f, 0.f, 0.f, 0.f};
    const float* arow0 = &qtile[col * QT_STRIDE + 2 * hi];         // A row = lane%16
    const float* arow1 = &qtile[(16 + col) * QT_STRIDE + 2 * hi];  // same bank pattern
#pragma unroll
    for (int c = 0; c < 32; ++c) {
      v2f a0 = *(const v2f*)(arow0 + 4 * c);  // conflict-free ds_load_b64
      v2f a1 = *(const v2f*)(arow1 + 4 * c);
      acc0 = __builtin_amdgcn_wmma_f32_16x16x4_f32(
          false, a0, false, bfrag[c], (short)0, acc0, false, false);
      acc1 = __builtin_amdgcn_wmma_f32_16x16x4_f32(
          false, a1, false, bfrag[c], (short)0, acc1, false, false);
    }

    // Running argmax; D layout: VGPR j -> query row base + j + 8*hi,
    // column = lane%16 (our key). Ascending n preserves first-occurrence ties.
#pragma unroll
    for (int j = 0; j < 8; ++j) {
      float val = acc0[j];
      int   n   = nbase + 8 * hi + j;
      if (val > bmax) { bmax = val; bidx = n; }
    }
#pragma unroll
    for (int j = 0; j < 8; ++j) {
      float val = acc1[j];
      int   n   = nbase + 16 + 8 * hi + j;
      if (val > bmax) { bmax = val; bidx = n; }
    }
  }

  // Merge the two half-wave partial argmaxes (same key, disjoint query rows).
  {
    float omax = __shfl_xor(bmax, 16, 32);
    int   oidx = __shfl_xor(bidx, 16, 32);
    if (omax > bmax || (omax == bmax && oidx < bidx)) { bmax = omax; bidx = oidx; }
  }

  // Scatter: both lanes holding key m add 64 elements each of v[b,m,:] into
  // out[b, bidx, :] (atomics land in 1MB L2-resident region).
  if (isfinite(bmax)) {
    const float* vrow = v   + ((size_t)b * M_ + m)    * D_ + hi * 64;
    float*       orow = out + ((size_t)b * N_ + bidx) * D_ + hi * 64;
#pragma unroll 8
    for (int d = 0; d < 64; ++d) atomic_add_f32(&orow[d], vrow[d]);
    if (!hi) atomic_add_f32(&counts[b * N_ + bidx], 1.0f);
  }
}

__global__ void normalize_kernel(float* __restrict__ out,
                                 const float* __restrict__ counts) {
  int i = blockIdx.x * blockDim.x + threadIdx.x;
  if (i >= B_ * N_ * D_) return;
  float c = counts[i / D_];
  out[i] = out[i] / (fmaxf(c, 1.0f) + 1e-6f);
}

extern "C" void kernel_launch(void* const* d_in, const int* in_sizes, int n_in,
                              void* d_out, int out_size, void* d_ws, size_t ws_size,
                              hipStream_t stream) {
  const float* q = (const float*)d_in[0];
  const float* k = (const float*)d_in[1];
  const float* v = (const float*)d_in[2];
  float* out    = (float*)d_out;
  float* counts = (float*)d_ws;  // B*N floats = 8KB scratch

  const int zn = B_ * N_ * D_;
  zero_kernel<<<(zn + 255) / 256, 256, 0, stream>>>(out, counts);

  dim3 grid(M_ / KEYS_PER_BLOCK, B_);  // (512, 2)
  argmax_scatter_kernel<<<grid, 256, 0, stream>>>(q, k, v, out, counts);

  normalize_kernel<<<(zn + 255) / 256, 256, 0, stream>>>(out, counts);
}